// PredictionCapsule_40157944218425
// MI455X (gfx1250) — compile-verified
//
#include <hip/hip_runtime.h>

// Capsule dynamic routing, MI455X (gfx1250, wave32, WMMA).
//
// Roofline: u=[64,32,2048,16] f32 = 268MB in HBM, re-read 2x/iter (~1.9GB,
// ~80us at 23.3 TB/s). Instead recompute u each routing pass with
// V_WMMA_F32_16X16X4_F32 (~1 GFLOP/pass); x (4MB) + W (33.5MB) stay resident
// in the 192MB L2. Workspace: logits 16MB + s/v 128KB each.
//
// Tile mapping: one 16x16 WMMA output tile == u[b0:16, c, i, 0:16]
//   A (16x4 f32) = x[b0:16, i, f..],  B (4x16) = W[c,i,d,f..]^T, K=8 -> 2 WMMAs.
// Workgroup = 32 waves (one per output capsule c) so softmax over c is an LDS
// exchange. IB=4 input capsules are processed per barrier phase. Agreement
// reduction over d uses DS_SWIZZLE_B32 xor patterns (no index VALU).

#define Bc   64
#define NIc  2048
#define DIc  8
#define NCc  32
#define DCc  16
#define ICH  64
#define IB   4

typedef __attribute__((ext_vector_type(2))) float v2f;
typedef __attribute__((ext_vector_type(8))) float v8f;

// Sum across each 16-lane half via ds_swizzle xor patterns (offset[15]=0,
// and_mask=0x1F, xor_mask in [14:10]); xor {1,2,4,8} stays within the half.
__device__ __forceinline__ float half16_sum(float x) {
  x += __int_as_float(__builtin_amdgcn_ds_swizzle(__float_as_int(x), 0x041F));
  x += __int_as_float(__builtin_amdgcn_ds_swizzle(__float_as_int(x), 0x081F));
  x += __int_as_float(__builtin_amdgcn_ds_swizzle(__float_as_int(x), 0x101F));
  x += __int_as_float(__builtin_amdgcn_ds_swizzle(__float_as_int(x), 0x201F));
  return x;
}

// ITER==0: uniform coupling 1/NC, no agreement, no logit I/O.
// ITER==1: blog = NI*agree(u, v); store for pass 2; softmax; accumulate s.
// ITER==2: blog = load(blog) + NI*agree(u, v); softmax; accumulate s.
template <int ITER>
__global__ __launch_bounds__(1024)
void caps_pass(const float* __restrict__ x, const float* __restrict__ W,
               const float* __restrict__ v, float* __restrict__ blog,
               float* __restrict__ s) {
  __shared__ float x_lds[ICH][16][DIc];        // 32 KB shared A operand
  __shared__ float blog_lds[IB][NCc][16];      // 8 KB logits for softmax
  __shared__ float m_lds[IB][16];              // softmax max per (j,b)
  __shared__ float z_lds[IB][16];              // softmax denom per (j,b)

  const int lane = threadIdx.x;                // 0..31
  const int c    = threadIdx.y;                // 0..31: this wave's capsule
  const int b0   = blockIdx.y * 16;            // batch tile
  const int i0   = blockIdx.x * ICH;           // input-capsule chunk

  // Cooperatively stage x[b0:16, i0:i0+ICH, 0:8] into LDS.
  {
    const int t = c * 32 + lane;
    #pragma unroll
    for (int k = 0; k < (ICH * 16 * DIc) / 1024; ++k) {
      int idx = t + k * 1024;
      int f   = idx & (DIc - 1);
      int bl  = (idx >> 3) & 15;
      int ii  = idx >> 7;
      x_lds[ii][bl][f] = x[((size_t)(b0 + bl) * NIc + (i0 + ii)) * DIc + f];
    }
  }
  __syncthreads();

  const int nloc = lane & 15;                  // tile N index (d; b_local for A)
  const int half = lane >> 4;                  // lane half
  const int kb   = half * 2;                   // K pair base (16x16x4 layout)

  // v[b, c, d] for agreement; invariant over i.
  float vv[8];
  if (ITER >= 1) {
    #pragma unroll
    for (int r = 0; r < 8; ++r) {
      int b = b0 + r + 8 * half;
      vv[r] = v[((size_t)b * NCc + c) * DCc + nloc];
    }
  }

  v8f sacc = {};                               // partial s tile (16b x 16d / wave)

  for (int ib = 0; ib < ICH; ib += IB) {
    // Prefetch next W slab (speculative; OOB prefetches are dropped).
    __builtin_prefetch(
        W + (((size_t)c * NIc + (i0 + ib + IB)) * DCc + nloc) * DIc, 0, 1);

    // ---- GEMM phase: IB u-tiles via WMMA ----
    v8f acc[IB];
    #pragma unroll
    for (int j = 0; j < IB; ++j) {
      const int ii = ib + j;
      const int i  = i0 + ii;
      v2f a0, a1;
      a0.x = x_lds[ii][nloc][kb + 0];  a0.y = x_lds[ii][nloc][kb + 1];
      a1.x = x_lds[ii][nloc][kb + 4];  a1.y = x_lds[ii][nloc][kb + 5];
      const float* wp = W + (((size_t)c * NIc + i) * DCc + nloc) * DIc;
      v2f w0, w1;
      w0.x = wp[kb + 0];  w0.y = wp[kb + 1];
      w1.x = wp[kb + 4];  w1.y = wp[kb + 5];
      v8f t = {};
      t = __builtin_amdgcn_wmma_f32_16x16x4_f32(false, a0, false, w0,
                                                (short)0, t, false, false);
      t = __builtin_amdgcn_wmma_f32_16x16x4_f32(false, a1, false, w1,
                                                (short)0, t, false, false);
      acc[j] = t;   // acc[j][r] = u[b0 + r + 8*half, c, i0+ib+j, nloc]
    }

    if (ITER == 0) {
      #pragma unroll
      for (int j = 0; j < IB; ++j)
        #pragma unroll
        for (int r = 0; r < 8; ++r) sacc[r] += (1.0f / NCc) * acc[j][r];
    } else {
      // ---- agreement + logit update (branch-free lane ownership) ----
      float blv[IB];
      #pragma unroll
      for (int j = 0; j < IB; ++j) {
        blv[j] = 0.0f;
        #pragma unroll
        for (int r = 0; r < 8; ++r) {
          float ag = half16_sum(acc[j][r] * vv[r]);   // sum over d
          float bl_new = (float)NIc * ag;             // NI * <u, v>
          blv[j] = ((nloc & 7) == r) ? bl_new : blv[j];
        }
      }
      // lanes 0-7 own b=0..7, lanes 16-23 own b=8..15
      if (nloc < 8) {
        const int b = (nloc & 7) + 8 * half;
        #pragma unroll
        for (int j = 0; j < IB; ++j) {
          size_t bidx = ((size_t)(b0 + b) * NCc + c) * NIc + (i0 + ib + j);
          if (ITER == 2) blv[j] += blog[bidx];   // accumulate prior logits
          else           blog[bidx] = blv[j];    // persist for next pass
          blog_lds[j][c][b] = blv[j];
        }
      }
      __syncthreads();

      // ---- softmax stats over c: wave 0, 2 (j,b) entries per lane ----
      if (c == 0) {
        #pragma unroll
        for (int e = lane; e < IB * 16; e += 32) {
          const int j = e >> 4, b = e & 15;
          float mx = -3.0e38f;
          for (int cc = 0; cc < NCc; ++cc) mx = fmaxf(mx, blog_lds[j][cc][b]);
          float zs = 0.0f;
          for (int cc = 0; cc < NCc; ++cc) zs += __expf(blog_lds[j][cc][b] - mx);
          m_lds[j][b] = mx;
          z_lds[j][b] = zs;
        }
      }
      __syncthreads();

      // ---- weighted accumulation into s partials ----
      #pragma unroll
      for (int j = 0; j < IB; ++j)
        #pragma unroll
        for (int r = 0; r < 8; ++r) {
          const int b = r + 8 * half;
          float coup = __expf(blog_lds[j][c][b] - m_lds[j][b]) / z_lds[j][b];
          sacc[r] += coup * acc[j][r];
        }
      __syncthreads();   // blog_lds reused next batch
    }
  }

  // s[b, c, d] += partials (s zeroed before each pass)
  #pragma unroll
  for (int r = 0; r < 8; ++r) {
    int b = b0 + r + 8 * half;
    atomicAdd(&s[((size_t)b * NCc + c) * DCc + nloc], sacc[r]);
  }
}

// v = squash(s) per (b,c); re-zeroes s for the next pass's atomics.
__global__ void caps_squash(float* __restrict__ s, float* __restrict__ v,
                            float* __restrict__ out) {
  int idx = blockIdx.x * blockDim.x + threadIdx.x;   // one thread per (b,c)
  if (idx >= Bc * NCc) return;
  float sv[DCc];
  float sq = 0.0f;
  float* sp = s + (size_t)idx * DCc;
  #pragma unroll
  for (int d = 0; d < DCc; ++d) {
    sv[d] = sp[d];
    sq += sv[d] * sv[d];
    sp[d] = 0.0f;
  }
  float scale = (sq / (1.0f + sq)) * rsqrtf(sq + 1e-7f);
  float* vp = v + (size_t)idx * DCc;
  #pragma unroll
  for (int d = 0; d < DCc; ++d) {
    float vd = sv[d] * scale;
    vp[d] = vd;
    if (out) out[(size_t)idx * DCc + d] = vd;
  }
}

extern "C" void kernel_launch(void* const* d_in, const int* in_sizes, int n_in,
                              void* d_out, int out_size, void* d_ws, size_t ws_size,
                              hipStream_t stream) {
  const float* x = (const float*)d_in[0];   // [64, 2048, 8]
  const float* W = (const float*)d_in[1];   // [32, 2048, 16, 8]
  float* out = (float*)d_out;               // [64, 32, 16]

  char* ws = (char*)d_ws;
  float* s    = (float*)ws;                        // 128 KB
  float* v    = (float*)(ws + 131072);             // 128 KB
  float* blog = (float*)(ws + 262144);             // 16 MB logits [B,NC,NI]

  hipMemsetAsync(s, 0, (size_t)Bc * NCc * DCc * sizeof(float), stream);

  dim3 blk(32, 32);                 // 32 waves: one per output capsule
  dim3 grd(NIc / ICH, Bc / 16);     // 32 x 4 workgroups

  caps_pass<0><<<grd, blk, 0, stream>>>(x, W, v, blog, s);
  caps_squash<<<(Bc * NCc + 127) / 128, 128, 0, stream>>>(s, v, nullptr);
  caps_pass<1><<<grd, blk, 0, stream>>>(x, W, v, blog, s);
  caps_squash<<<(Bc * NCc + 127) / 128, 128, 0, stream>>>(s, v, nullptr);
  caps_pass<2><<<grd, blk, 0, stream>>>(x, W, v, blog, s);
  caps_squash<<<(Bc * NCc + 127) / 128, 128, 0, stream>>>(s, v, out);
}